// GCN_4_layer_fc_45311904973175
// MI455X (gfx1250) — compile-verified
//
#include <hip/hip_runtime.h>
#include <hip/hip_bf16.h>

typedef __attribute__((ext_vector_type(2))) float v2f;
typedef __attribute__((ext_vector_type(8))) float v8f;

// ---------------------------------------------------------------------------
// Utility kernels
// ---------------------------------------------------------------------------
__global__ void fill_zero_kernel(float* __restrict__ p, long n) {
  long i = (long)blockIdx.x * blockDim.x + threadIdx.x;
  if (i < n) p[i] = 0.0f;
}

__global__ void degree_kernel(const int* __restrict__ src, const int* __restrict__ dst,
                              float* __restrict__ deg_out, float* __restrict__ deg_in, int E) {
  int e = blockIdx.x * blockDim.x + threadIdx.x;
  if (e < E) {
    atomicAdd(&deg_out[src[e]], 1.0f);
    atomicAdd(&deg_in[dst[e]], 1.0f);
  }
}

__global__ void norm_kernel(const float* __restrict__ deg_out, const float* __restrict__ deg_in,
                            float* __restrict__ src_norm, float* __restrict__ dst_norm, int n) {
  int i = blockIdx.x * blockDim.x + threadIdx.x;
  if (i < n) {
    src_norm[i] = rsqrtf(fmaxf(deg_out[i], 1.0f));
    dst_norm[i] = rsqrtf(fmaxf(deg_in[i], 1.0f));
  }
}

// ---------------------------------------------------------------------------
// WMMA GEMM: out[m,n] = (sum_k X[m,k]*W[k,n]) * rowscale[m]? + bias[n]? (relu?)
// One wave32 per 16x16 output tile. blockDim = (32, NT/16), gridDim.x = row tiles.
// Uses V_WMMA_F32_16X16X4_F32 (fp32 A/B/C, K=4 per instruction).
//
// Fragment layouts (CDNA5 ISA 7.12.2):
//   A 16x4 : lanes 0-15 row M=lane  hold K = {k0,k0+1}; lanes 16-31 hold {k0+2,k0+3}
//   B 4x16 : VGPR0 lanes 0-15 -> K=k0   N=lane ; lanes 16-31 -> K=k0+2 N=lane-16
//            VGPR1               K=k0+1         ;               K=k0+3
//   C 16x16: VGPR v, lanes 0-15 -> M=v, N=lane ; lanes 16-31 -> M=v+8, N=lane-16
// ---------------------------------------------------------------------------
template <int K, int NT, bool HAS_SCALE, bool HAS_BIAS, bool RELU>
__global__ void wmma_gemm_kernel(const float* __restrict__ X, const float* __restrict__ W,
                                 const float* __restrict__ rowscale,
                                 const float* __restrict__ bias,
                                 float* __restrict__ out, int nrows) {
  const int lane    = threadIdx.x;       // 0..31
  const int colTile = threadIdx.y;       // 0..NT/16-1
  const int rowTile = blockIdx.x;
  const int hh      = lane >> 4;         // which half-wave
  const int l16     = lane & 15;

  int m = rowTile * 16 + l16;            // A row handled by this lane
  if (m > nrows - 1) m = nrows - 1;      // clamp loads; keep EXEC all-ones for WMMA
  const int n = colTile * 16 + l16;      // B/C column handled by this lane

  const float* __restrict__ Xrow = X + (size_t)m * K;

  v8f c = {};
#pragma unroll
  for (int k0 = 0; k0 < K; k0 += 4) {
    const int ka = k0 + hh * 2;
    v2f a;
    a.x = Xrow[ka];
    a.y = Xrow[ka + 1];
    v2f b;
    b.x = W[(size_t)ka * NT + n];
    b.y = W[(size_t)(ka + 1) * NT + n];
    c = __builtin_amdgcn_wmma_f32_16x16x4_f32(
        /*neg_a=*/false, a, /*neg_b=*/false, b,
        /*c_mod=*/(short)0, c, /*reuse_a=*/false, /*reuse_b=*/false);
  }

  const float bn = HAS_BIAS ? bias[n] : 0.0f;
#pragma unroll
  for (int v = 0; v < 8; ++v) {
    const int mr = rowTile * 16 + v + 8 * hh;   // C row for element v
    if (mr < nrows) {
      float val = c[v];
      if (HAS_SCALE) val *= rowscale[mr];
      if (HAS_BIAS) val += bn;
      if (RELU) val = fmaxf(val, 0.0f);
      out[(size_t)mr * NT + n] = val;
    }
  }
}

// ---------------------------------------------------------------------------
// Edge scatter-add: agg[dst[e], :] += hs[src[e], :]   (64 features/edge)
// 64 consecutive threads cooperate on one edge -> one coalesced 256B read,
// 64 global_atomic_add_f32 into L2-resident agg (25.6MB << 192MB L2).
// ---------------------------------------------------------------------------
__global__ void scatter_add_kernel(const float* __restrict__ hs, const int* __restrict__ src,
                                   const int* __restrict__ dst, float* __restrict__ agg, int E) {
  const int f = threadIdx.x & 63;
  long group = ((long)blockIdx.x * blockDim.x + threadIdx.x) >> 6;
  const long nGroups = ((long)gridDim.x * blockDim.x) >> 6;
  for (long e = group; e < E; e += nGroups) {
    const int s = src[e];
    const int d = dst[e];
    atomicAdd(&agg[(size_t)d * 64 + f], hs[(size_t)s * 64 + f]);
  }
}

// out = relu(agg * dst_norm[m] + b[n])
__global__ void finalize_relu_kernel(const float* __restrict__ agg,
                                     const float* __restrict__ dst_norm,
                                     const float* __restrict__ bias,
                                     float* __restrict__ out, long total) {
  long i = (long)blockIdx.x * blockDim.x + threadIdx.x;
  if (i < total) {
    const long m = i >> 6;
    const int nn = (int)(i & 63);
    const float v = agg[i] * dst_norm[m] + bias[nn];
    out[i] = fmaxf(v, 0.0f);
  }
}

// out = relu(agg * dst_norm[m] + b[n] + res)   (layer-4 + residual merge)
__global__ void finalize_res_kernel(const float* __restrict__ agg,
                                    const float* __restrict__ dst_norm,
                                    const float* __restrict__ bias,
                                    const float* __restrict__ res,
                                    float* __restrict__ out, long total) {
  long i = (long)blockIdx.x * blockDim.x + threadIdx.x;
  if (i < total) {
    const long m = i >> 6;
    const int nn = (int)(i & 63);
    const float v = agg[i] * dst_norm[m] + bias[nn] + res[i];
    out[i] = fmaxf(v, 0.0f);
  }
}

// ---------------------------------------------------------------------------
// Launch
// ---------------------------------------------------------------------------
extern "C" void kernel_launch(void* const* d_in, const int* in_sizes, int n_in,
                              void* d_out, int out_size, void* d_ws, size_t ws_size,
                              hipStream_t stream) {
  const float* x    = (const float*)d_in[0];   // [N,128]
  const int*   src  = (const int*)d_in[1];     // [E]
  const int*   dst  = (const int*)d_in[2];     // [E]
  const float* W1   = (const float*)d_in[3];   // [128,64]
  const float* b1   = (const float*)d_in[4];
  const float* W2   = (const float*)d_in[5];   // [64,64]
  const float* b2   = (const float*)d_in[6];
  const float* W3   = (const float*)d_in[7];
  const float* b3   = (const float*)d_in[8];
  const float* W4   = (const float*)d_in[9];
  const float* b4   = (const float*)d_in[10];
  const float* Wres = (const float*)d_in[11];  // [128,64]
  const float* bres = (const float*)d_in[12];
  const float* Wop  = (const float*)d_in[13];  // [64,16]
  const float* bop  = (const float*)d_in[14];

  const int N = in_sizes[0] / 128;
  const int E = in_sizes[1];

  // workspace layout (floats)
  float* ws       = (float*)d_ws;
  float* deg_out  = ws;                       // N
  float* deg_in   = deg_out + N;              // N
  float* src_norm = deg_in + N;               // N
  float* dst_norm = src_norm + N;             // N
  float* hs       = dst_norm + N;             // N*64 (scaled messages)
  float* agg      = hs + (size_t)N * 64;      // N*64 (scatter accumulator)
  float* out      = agg + (size_t)N * 64;     // N*64 (layer output)
  float* res      = out + (size_t)N * 64;     // N*64 (residual branch)

  const long NF = (long)N * 64;
  const dim3 gblk(32, 4);                     // 4 waves: one per 16-col tile
  const int rowTiles = (N + 15) / 16;

  // ---- degree + norms -------------------------------------------------
  fill_zero_kernel<<<(2 * (long)N + 255) / 256, 256, 0, stream>>>(deg_out, 2 * (long)N);
  degree_kernel<<<(E + 255) / 256, 256, 0, stream>>>(src, dst, deg_out, deg_in, E);
  norm_kernel<<<(N + 255) / 256, 256, 0, stream>>>(deg_out, deg_in, src_norm, dst_norm, N);

  // ---- residual branch: res = x @ Wres + bres -------------------------
  wmma_gemm_kernel<128, 64, false, true, false>
      <<<rowTiles, gblk, 0, stream>>>(x, Wres, nullptr, bres, res, N);

  // ---- layer 1 --------------------------------------------------------
  wmma_gemm_kernel<128, 64, true, false, false>
      <<<rowTiles, gblk, 0, stream>>>(x, W1, src_norm, nullptr, hs, N);
  fill_zero_kernel<<<(NF + 255) / 256, 256, 0, stream>>>(agg, NF);
  scatter_add_kernel<<<8192, 256, 0, stream>>>(hs, src, dst, agg, E);
  finalize_relu_kernel<<<(NF + 255) / 256, 256, 0, stream>>>(agg, dst_norm, b1, out, NF);

  // ---- layer 2 --------------------------------------------------------
  wmma_gemm_kernel<64, 64, true, false, false>
      <<<rowTiles, gblk, 0, stream>>>(out, W2, src_norm, nullptr, hs, N);
  fill_zero_kernel<<<(NF + 255) / 256, 256, 0, stream>>>(agg, NF);
  scatter_add_kernel<<<8192, 256, 0, stream>>>(hs, src, dst, agg, E);
  finalize_relu_kernel<<<(NF + 255) / 256, 256, 0, stream>>>(agg, dst_norm, b2, out, NF);

  // ---- layer 3 --------------------------------------------------------
  wmma_gemm_kernel<64, 64, true, false, false>
      <<<rowTiles, gblk, 0, stream>>>(out, W3, src_norm, nullptr, hs, N);
  fill_zero_kernel<<<(NF + 255) / 256, 256, 0, stream>>>(agg, NF);
  scatter_add_kernel<<<8192, 256, 0, stream>>>(hs, src, dst, agg, E);
  finalize_relu_kernel<<<(NF + 255) / 256, 256, 0, stream>>>(agg, dst_norm, b3, out, NF);

  // ---- layer 4 (+ residual, relu) ------------------------------------
  wmma_gemm_kernel<64, 64, true, false, false>
      <<<rowTiles, gblk, 0, stream>>>(out, W4, src_norm, nullptr, hs, N);
  fill_zero_kernel<<<(NF + 255) / 256, 256, 0, stream>>>(agg, NF);
  scatter_add_kernel<<<8192, 256, 0, stream>>>(hs, src, dst, agg, E);
  finalize_res_kernel<<<(NF + 255) / 256, 256, 0, stream>>>(agg, dst_norm, b4, res, out, NF);

  // ---- head: d_out = out @ Wop + bop  (N x 16) ------------------------
  wmma_gemm_kernel<64, 16, false, true, false>
      <<<rowTiles, dim3(32, 1), 0, stream>>>(out, Wop, nullptr, bop, (float*)d_out, N);
}